// DownSampling_58961311040322
// MI455X (gfx1250) — compile-verified
//
#include <hip/hip_runtime.h>
#include <hip/hip_bf16.h>
#include <stdint.h>

#define BATCH 8
#define NPTS  32768
#define CINCH 128
#define MOUT  (NPTS / 4)       // 8192
#define TPB   1024
#define PPT   (NPTS / TPB)     // 32 points per thread

// Dynamic LDS layout (floats):
//  [0 .. 2N)         : x row then y row (staged by TDM)
//  [2N .. 2N+64)     : double-buffered per-wave partial best distance  [2][32]
//  [2N+64 .. +128)   : double-buffered per-wave partial best z         [2][32]
//  [2N+128 .. +192)  : double-buffered per-wave partial best index     [2][32]
//  [2N+192]          : init-time broadcast of z(point 0)
#define SMEM_FLOATS (2 * NPTS + 193)
#define SMEM_BYTES  (SMEM_FLOATS * 4)

typedef unsigned int u32x4 __attribute__((ext_vector_type(4)));
typedef int          i32x8 __attribute__((ext_vector_type(8)));
typedef int          i32x4 __attribute__((ext_vector_type(4)));

// One TDM descriptor: load a 32768x2 f32 tile (x row + y row, row stride N)
// from global into LDS at lds_off. 256KB moved by one instruction,
// tracked by TENSORcnt.
__device__ __forceinline__ void tdm_load_xy_to_lds(const float* gsrc, uint32_t lds_off) {
  uint64_t ga = (uint64_t)(uintptr_t)gsrc;
  u32x4 g0;
  g0[0] = 1u;                                              // count=1 (valid), user mode
  g0[1] = lds_off;                                         // LDS byte address
  g0[2] = (uint32_t)(ga & 0xffffffffu);                    // global_addr[31:0]
  g0[3] = (uint32_t)((ga >> 32) & 0x1ffffffu) | (2u << 30);// global_addr[56:32] | type=2
  i32x8 g1;
  g1[0] = (int)(2u << 16);        // workgroup_mask=0, data_size=2 (4B), no barrier/iter/pad
  g1[1] = (int)(0x8000u << 16);   // tensor_dim0[15:0]=32768 in [31:16]
  g1[2] = (int)(3u << 16);        // tensor_dim0[31:16]=0 | tensor_dim1[15:0]=3 (x,y,z rows)
  g1[3] = (int)(0x8000u << 16);   // tensor_dim1 hi=0 | tile_dim0=32768
  g1[4] = 2;                      // tile_dim1=2 (x and y), tile_dim2=0
  g1[5] = (int)NPTS;              // tensor_dim0_stride[31:0] = 32768
  g1[6] = 0;                      // stride hi | dim1_stride lo
  g1[7] = 0;
  i32x4 z4 = {0, 0, 0, 0};        // groups 2/3 unused (<=2D tile)
  i32x8 z8 = {0, 0, 0, 0, 0, 0, 0, 0};
  __builtin_amdgcn_tensor_load_to_lds(g0, g1, z4, z4, z8, 0);
}

__global__ __launch_bounds__(TPB, 1) void fps_kernel(const float* __restrict__ xyz,
                                                     int* __restrict__ idx_out) {
  extern __shared__ float smem[];
  float* sxy   = smem;                 // [2*NPTS] x then y
  float* s_pd  = smem + 2 * NPTS;      // [2][32]
  float* s_pz  = s_pd + 64;            // [2][32]
  int*   s_pi  = (int*)(s_pz + 64);    // [2][32]
  float* s_cz0 = (float*)(s_pi + 64);  // init-only broadcast

  const int b    = blockIdx.x;
  const int tid  = threadIdx.x;
  const int lane = tid & 31;
  const int wave = tid >> 5;
  const float* base = xyz + (size_t)b * 3 * NPTS;

  // z coords + running min-distance live in registers: 32 points/thread.
  float z[PPT], dist[PPT];
#pragma unroll
  for (int k = 0; k < PPT; ++k) {
    z[k]    = base[2 * NPTS + tid + (k << 10)];
    dist[k] = 1e10f;
  }

  if (tid == 0) {
    tdm_load_xy_to_lds(base, (uint32_t)(uintptr_t)sxy);
    __builtin_amdgcn_s_wait_tensorcnt(0);
    *s_cz0 = z[0];                 // thread 0 owns point 0
  }
  __syncthreads();

  int   far = 0;                   // uniform across all threads
  float cz  = *s_cz0;

  for (int m = 0; m < MOUT; ++m) {
    if (tid == 0) idx_out[b * MOUT + m] = far;
    const float cx = sxy[far];
    const float cy = sxy[NPTS + far];

    // Update distances, track local argmax (strict > keeps lowest index).
    float bd = -1.0f;
    int   bi = 0;
    float bz = 0.0f;
#pragma unroll
    for (int k = 0; k < PPT; ++k) {
      const int   n   = tid + (k << 10);
      const float dx  = sxy[n] - cx;
      const float dy  = sxy[NPTS + n] - cy;
      const float dzv = z[k] - cz;
      float d = fmaf(dzv, dzv, fmaf(dy, dy, dx * dx));
      d = fminf(dist[k], d);
      dist[k] = d;
      if (d > bd) { bd = d; bi = n; bz = z[k]; }
    }

    // wave32 xor-butterfly reduction carrying (dist, idx, z); ties -> lower index.
#pragma unroll
    for (int off = 16; off > 0; off >>= 1) {
      const float od = __shfl_xor(bd, off, 32);
      const int   oi = __shfl_xor(bi, off, 32);
      const float oz = __shfl_xor(bz, off, 32);
      if (od > bd || (od == bd && oi < bi)) { bd = od; bi = oi; bz = oz; }
    }

    const int buf = (m & 1) << 5;     // double-buffered partials -> 1 barrier/iter
    if (lane == 0) {
      s_pd[buf + wave] = bd;
      s_pi[buf + wave] = bi;
      s_pz[buf + wave] = bz;
    }
    __syncthreads();

    // Every wave redundantly reduces the 32 wave partials; xor butterfly leaves
    // the identical winner in every lane of every wave -> no second barrier,
    // no wave-0 serialization.
    bd = s_pd[buf + lane];
    bi = s_pi[buf + lane];
    bz = s_pz[buf + lane];
#pragma unroll
    for (int off = 16; off > 0; off >>= 1) {
      const float od = __shfl_xor(bd, off, 32);
      const int   oi = __shfl_xor(bi, off, 32);
      const float oz = __shfl_xor(bz, off, 32);
      if (od > bd || (od == bd && oi < bi)) { bd = od; bi = oi; bz = oz; }
    }
    far = bi;
    cz  = bz;
  }
}

// One block per (batch, channel) row: coalesced idx loads + writes,
// gather reads land in a 128KB L2-resident row.
__global__ void gather_kernel(const float* __restrict__ xyz,
                              const float* __restrict__ feat,
                              const int* __restrict__ idx,
                              float* __restrict__ out) {
  const int blk = blockIdx.x;                 // b*(3+CIN) + c
  const int b   = blk / (3 + CINCH);
  const int c   = blk % (3 + CINCH);
  const float* src;
  float* dst;
  if (c < 3) {
    src = xyz + ((size_t)b * 3 + c) * NPTS;
    dst = out + ((size_t)b * 3 + c) * MOUT;
  } else {
    const int cf = c - 3;
    src = feat + ((size_t)b * CINCH + cf) * NPTS;
    dst = out + (size_t)BATCH * 3 * MOUT + ((size_t)b * CINCH + cf) * MOUT;
  }
  const int* ib = idx + b * MOUT;
  for (int m = threadIdx.x; m < MOUT; m += blockDim.x) {
    dst[m] = src[ib[m]];
  }
}

extern "C" void kernel_launch(void* const* d_in, const int* in_sizes, int n_in,
                              void* d_out, int out_size, void* d_ws, size_t ws_size,
                              hipStream_t stream) {
  const float* xyz  = (const float*)d_in[0];   // [B,3,N]
  const float* feat = (const float*)d_in[1];   // [B,CIN,N]
  float* out = (float*)d_out;                  // [B,3,M] ++ [B,CIN,M]
  int*   idx = (int*)d_ws;                     // [B,M] scratch (256KB)

  fps_kernel<<<BATCH, TPB, SMEM_BYTES, stream>>>(xyz, idx);
  gather_kernel<<<BATCH * (3 + CINCH), 256, 0, stream>>>(xyz, feat, idx, out);
}